// Node_Transformation_76501957476874
// MI455X (gfx1250) — compile-verified
//
#include <hip/hip_runtime.h>

typedef __attribute__((ext_vector_type(2))) float v2f;
typedef __attribute__((ext_vector_type(4))) float v4f;
typedef __attribute__((ext_vector_type(8))) float v8f;

#define IN_CH   256
#define HID     128
#define PITCH2  144          // K-pair-major pitch in float2: %32==16 -> hi=0 and
                             // hi=1 halves hit disjoint bank sets; t*128B offsets
                             // fit the 16-bit DS immediate
#define WAVES_PER_BLOCK 16   // 512 threads; each wave does 32 rows
#define ROWS_PER_WAVE   32
#define LDS_BYTES ((IN_CH / 2) * PITCH2 * 8)   // 147456 B -> 2 blocks / 320KB WGP

// 8-column-tile WMMA K-loop over one or two 16-row halves.
// Each B fragment is ONE ds_load_b64 (contiguous float2 in LDS, immediate
// offset t*128B) -> lands directly in an even-aligned VGPR pair for WMMA.
template <int NSTRIP>
__device__ __forceinline__ void gemm_ktiles(const float* __restrict__ xp0,
                                            const float* __restrict__ xp1,
                                            const v2f* __restrict__ wl,
                                            v8f* acc0, v8f* acc1)
{
    #pragma unroll 2
    for (int k = 0; k < IN_CH; k += 4) {
        v2f a0 = *(const v2f*)(xp0 + k);              // global_load_b64
        v2f a1;
        if (NSTRIP == 2) a1 = *(const v2f*)(xp1 + k); // global_load_b64
        const v2f* wk = wl + (k >> 1) * PITCH2;       // one v_add per step
        #pragma unroll
        for (int t = 0; t < 8; ++t) {
            v2f b = wk[t * 16];                       // ds_load_b64, imm offset
            acc0[t] = __builtin_amdgcn_wmma_f32_16x16x4_f32(
                false, a0, false, b, (short)0, acc0[t], false, false);
            if (NSTRIP == 2)
                acc1[t] = __builtin_amdgcn_wmma_f32_16x16x4_f32(
                    false, a1, false, b, (short)0, acc1[t], false, false);
        }
    }
}

// One wave32 computes a 32-row x 128-col output strip:
//   lin = X[32x256] * W^T[256x128] via V_WMMA_F32_16X16X4_F32 (exact fp32).
// W is staged once per 16-wave workgroup into K-pair-major float2 LDS.
__global__ __launch_bounds__(512) void node_transform_wmma(
    const float* __restrict__ x,          // [N, 256]
    const int*   __restrict__ node_type,  // [N]
    const int*   __restrict__ item_id_p,  // [1]
    const float* __restrict__ emb,        // [8, 128]
    const float* __restrict__ W,          // [128, 256] row-major
    const float* __restrict__ bias,       // [128]
    float*       __restrict__ out,        // [N, 128]
    int N, int n_strips)
{
    extern __shared__ float wlds[];
    v2f* wl2 = (v2f*)wlds;                // wl2[k2 * PITCH2 + h] = W[h][2k2..2k2+1]

    // ---- Stage W -> LDS (transposed to K-pair-major). Consecutive lanes ->
    // consecutive h -> banks {2h,2h+1}: conflict-free ds_store_b64. W is 128KB,
    // read exactly once per block out of L2.
    for (int i = threadIdx.x; i < (IN_CH / 4) * HID; i += blockDim.x) {
        const int h  = i & (HID - 1);
        const int k4 = i >> 7;
        v4f wv = *(const v4f*)(W + h * IN_CH + k4 * 4);
        v2f lo; lo.x = wv.x; lo.y = wv.y;
        v2f hi2; hi2.x = wv.z; hi2.y = wv.w;
        wl2[(2 * k4 + 0) * PITCH2 + h] = lo;
        wl2[(2 * k4 + 1) * PITCH2 + h] = hi2;
    }
    __syncthreads();

    const int wave = blockIdx.x * WAVES_PER_BLOCK + (threadIdx.x >> 5);
    if (wave >= n_strips) return;          // wave-uniform: EXEC stays all-1s

    const int  lane = threadIdx.x & 31;
    const int  m    = lane & 15;           // A: row in half-tile / B: column
    const int  hi   = lane >> 4;           // K sub-pair select
    const long row_base = (long)wave * ROWS_PER_WAVE;
    const int  item = item_id_p[0];

    // One ballot covers all 32 rows of this wave (lane == row offset).
    long rchk = row_base + lane;
    if (rchk > (long)N - 1) rchk = (long)N - 1;
    const unsigned long long bal = __ballot(node_type[rchk] == item);
    const bool any0 = (bal & 0xFFFFull)     != 0;   // rows 0..15
    const bool any1 = (bal & 0xFFFF0000ull) != 0;   // rows 16..31

    v8f acc0[8] = {};
    v8f acc1[8] = {};

    long rm0 = row_base + m;        if (rm0 > (long)N - 1) rm0 = (long)N - 1;
    long rm1 = row_base + 16 + m;   if (rm1 > (long)N - 1) rm1 = (long)N - 1;
    const float* xp0 = x + rm0 * IN_CH + 2 * hi;
    const float* xp1 = x + rm1 * IN_CH + 2 * hi;
    // lane's B base: k2 = k/2 + hi, column m
    const v2f*   wl  = wl2 + hi * PITCH2 + m;

    if (any0 && any1)      gemm_ktiles<2>(xp0, xp1, wl, acc0, acc1);
    else if (any0)         gemm_ktiles<1>(xp0, xp0, wl, acc0, acc0);
    else if (any1)         gemm_ktiles<1>(xp1, xp1, wl, acc1, acc1);
    // neither: pure embedding gather below (acc stays zero, never selected)

    // D layout (16x16 f32): VGPR j -> row j (lanes 0-15) / j+8 (lanes 16-31),
    // lane&15 -> column. Per-row select between linear path and embedding.
    #pragma unroll
    for (int half = 0; half < 2; ++half) {
        const v8f*  acc = half ? acc1 : acc0;
        const long  rb  = row_base + half * 16;
        #pragma unroll
        for (int j = 0; j < 8; ++j) {
            const long r = rb + j + 8 * hi;
            if (r < N) {
                const int    nt       = node_type[r];
                const bool   take_lin = (nt == item);
                const float* erow     = emb + (long)nt * HID;
                #pragma unroll
                for (int t = 0; t < 8; ++t) {
                    const int c = t * 16 + m;
                    const float v = take_lin ? (acc[t][j] + bias[c]) : erow[c];
                    out[r * HID + c] = v;
                }
            }
        }
    }
}

extern "C" void kernel_launch(void* const* d_in, const int* in_sizes, int n_in,
                              void* d_out, int out_size, void* d_ws, size_t ws_size,
                              hipStream_t stream) {
    const float* x    = (const float*)d_in[0];
    const int*   nt   = (const int*)  d_in[1];
    const int*   item = (const int*)  d_in[2];
    const float* emb  = (const float*)d_in[3];
    const float* W    = (const float*)d_in[4];
    const float* b    = (const float*)d_in[5];
    float*       out  = (float*)d_out;

    const int N        = in_sizes[1];                    // rows
    const int n_strips = (N + ROWS_PER_WAVE - 1) / ROWS_PER_WAVE;
    const int blocks   = (n_strips + WAVES_PER_BLOCK - 1) / WAVES_PER_BLOCK;

    node_transform_wmma<<<blocks, WAVES_PER_BLOCK * 32, LDS_BYTES, stream>>>(
        x, nt, item, emb, W, b, out, N, n_strips);
}